// BilateralSlice_8761733284299
// MI455X (gfx1250) — compile-verified
//
#include <hip/hip_runtime.h>

// Problem constants (fixed by setup_inputs in the reference)
#define GD 8
#define GH 16
#define GW 16
#define NC 12                       // n_out*(n_in+1) = 12 coeff channels
#define IMG_H 1024
#define IMG_W 1024
#define HW (IMG_H * IMG_W)
#define GRID_ELEMS (NC * GD * GH * GW)   // 24576 floats = 96 KB
#define THREADS 256
#define PPT 16
#define PPB (THREADS * PPT)              // 4096 pixels per block
#define STAGE_ITERS (GRID_ELEMS / THREADS)   // 96

typedef __attribute__((address_space(1))) int glb_i32_t;
typedef __attribute__((address_space(3))) int lds_i32_t;

__device__ __forceinline__ int iclamp(int v, int lo, int hi) {
    return v < lo ? lo : (v > hi ? hi : v);
}

__global__ __launch_bounds__(THREADS)
void bilateral_slice_apply_kernel(const float* __restrict__ grid,
                                  const float* __restrict__ guide,
                                  const float* __restrict__ image,
                                  float* __restrict__ out) {
    // Channel-last grid tile: [z][y][x][c], c contiguous (48 B per cell)
    __shared__ float sgrid[GRID_ELEMS];

    const int tid  = threadIdx.x;
    const int nblk_per_img = HW / PPB;            // 256
    const int n    = blockIdx.x / nblk_per_img;
    const int chnk = blockIdx.x % nblk_per_img;
    const int pix_base = chnk * PPB;

    // ---- Stage grid[n] into LDS with transpose [c][zyx] -> [zyx][c] ----
    // Uniform counted loop: e = i*256 + tid. Since 256 | 2048, within each
    // unrolled iteration c = i>>3 and r = (i&7)*256 + tid are constants+tid,
    // so all 96 async loads share two base addresses + immediate offsets.
    const float* gsrc = grid + (size_t)n * GRID_ELEMS;
#if __has_builtin(__builtin_amdgcn_global_load_async_to_lds_b32)
#pragma unroll
    for (int i = 0; i < STAGE_ITERS; ++i) {
        const int c    = i >> 3;           // channel (uniform constant)
        const int rblk = (i & 7) * 256;    // zyx block base (uniform constant)
        __builtin_amdgcn_global_load_async_to_lds_b32(
            (glb_i32_t*)(gsrc + i * THREADS + tid),
            (lds_i32_t*)(sgrid + (rblk + tid) * NC + c),
            0, 0);
    }
#if __has_builtin(__builtin_amdgcn_s_wait_asynccnt)
    __builtin_amdgcn_s_wait_asynccnt(0);
#else
    asm volatile("s_wait_asynccnt 0" ::: "memory");
#endif
#else
#pragma unroll 4
    for (int i = 0; i < STAGE_ITERS; ++i) {
        const int c    = i >> 3;
        const int rblk = (i & 7) * 256;
        sgrid[(rblk + tid) * NC + c] = gsrc[i * THREADS + tid];
    }
#endif
    __syncthreads();

    const float* guide_n = guide + (size_t)n * HW;
    const float* img_n   = image + (size_t)n * 3 * HW;
    float*       out_n   = out   + (size_t)n * 3 * HW;

    const float sx = (float)GW / (float)IMG_W;    // 1/64
    const float sy = (float)GH / (float)IMG_H;    // 1/64

#pragma unroll 2
    for (int it = 0; it < PPT; ++it) {
        const int p = pix_base + tid + it * THREADS;
        const int y = p >> 10;            // p / IMG_W
        const int x = p & (IMG_W - 1);

        const float gv  = guide_n[p];
        const float gxf = ((float)x + 0.5f) * sx - 0.5f;
        const float gyf = ((float)y + 0.5f) * sy - 0.5f;
        const float gzf = gv * (float)GD - 0.5f;

        const float fxf = floorf(gxf), fyf = floorf(gyf), fzf = floorf(gzf);
        const float wx1 = gxf - fxf, wy1 = gyf - fyf, wz1 = gzf - fzf;
        const float wx0 = 1.0f - wx1, wy0 = 1.0f - wy1, wz0 = 1.0f - wz1;

        const int ix0 = iclamp((int)fxf,     0, GW - 1);
        const int ix1 = iclamp((int)fxf + 1, 0, GW - 1);
        const int iy0 = iclamp((int)fyf,     0, GH - 1);
        const int iy1 = iclamp((int)fyf + 1, 0, GH - 1);
        const int iz0 = iclamp((int)fzf,     0, GD - 1);
        const int iz1 = iclamp((int)fzf + 1, 0, GD - 1);

        // float offsets into sgrid ([z][y][x][c])
        const int zo0 = iz0 * (GH * GW * NC), zo1 = iz1 * (GH * GW * NC);
        const int yo0 = iy0 * (GW * NC),      yo1 = iy1 * (GW * NC);
        const int xo0 = ix0 * NC,             xo1 = ix1 * NC;

        float acc[NC];
#pragma unroll
        for (int k = 0; k < NC; ++k) acc[k] = 0.0f;

        auto corner = [&](int off, float w) {
            const float4* cp = (const float4*)(sgrid + off);  // 16B aligned (48B cells)
            const float4 a = cp[0];
            const float4 b = cp[1];
            const float4 c = cp[2];
            acc[0]  += w * a.x;  acc[1]  += w * a.y;
            acc[2]  += w * a.z;  acc[3]  += w * a.w;
            acc[4]  += w * b.x;  acc[5]  += w * b.y;
            acc[6]  += w * b.z;  acc[7]  += w * b.w;
            acc[8]  += w * c.x;  acc[9]  += w * c.y;
            acc[10] += w * c.z;  acc[11] += w * c.w;
        };

        corner(zo0 + yo0 + xo0, wz0 * wy0 * wx0);
        corner(zo0 + yo0 + xo1, wz0 * wy0 * wx1);
        corner(zo0 + yo1 + xo0, wz0 * wy1 * wx0);
        corner(zo0 + yo1 + xo1, wz0 * wy1 * wx1);
        corner(zo1 + yo0 + xo0, wz1 * wy0 * wx0);
        corner(zo1 + yo0 + xo1, wz1 * wy0 * wx1);
        corner(zo1 + yo1 + xo0, wz1 * wy1 * wx0);
        corner(zo1 + yo1 + xo1, wz1 * wy1 * wx1);

        // per-pixel affine apply: out_o = sum_i acc[o*4+i]*img_i + acc[o*4+3]
        const float i0 = img_n[p];
        const float i1 = img_n[p + HW];
        const float i2 = img_n[p + 2 * HW];

        const float o0 = acc[0] * i0 + acc[1] * i1 + acc[2]  * i2 + acc[3];
        const float o1 = acc[4] * i0 + acc[5] * i1 + acc[6]  * i2 + acc[7];
        const float o2 = acc[8] * i0 + acc[9] * i1 + acc[10] * i2 + acc[11];

        out_n[p]          = o0;
        out_n[p + HW]     = o1;
        out_n[p + 2 * HW] = o2;
    }
}

extern "C" void kernel_launch(void* const* d_in, const int* in_sizes, int n_in,
                              void* d_out, int out_size, void* d_ws, size_t ws_size,
                              hipStream_t stream) {
    const float* grid  = (const float*)d_in[0];   // [N,12,8,16,16]
    const float* guide = (const float*)d_in[1];   // [N,1024,1024]
    const float* image = (const float*)d_in[2];   // [N,3,1024,1024]
    float*       out   = (float*)d_out;           // [N,3,1024,1024]

    const int N = in_sizes[1] / HW;               // 4
    dim3 blocks(N * (HW / PPB));                  // 1024
    bilateral_slice_apply_kernel<<<blocks, THREADS, 0, stream>>>(grid, guide, image, out);
}